// GoSpecificWattiPooling_4458176053252
// MI455X (gfx1250) — compile-verified
//
#include <hip/hip_runtime.h>
#include <hip/hip_bf16.h>

typedef __attribute__((ext_vector_type(2))) float v2f;
typedef __attribute__((ext_vector_type(8))) float v8f;

constexpr int B_ = 8, L_ = 8192, T_ = 512, DH_ = 1024, DG_ = 768, DP_ = 256;

__device__ __forceinline__ v8f wmma4(v2f a, v2f b, v8f c) {
  // V_WMMA_F32_16X16X4_F32 : D = A(16x4) * B(4x16) + C(16x16), fp32
  return __builtin_amdgcn_wmma_f32_16x16x4_f32(
      /*neg_a=*/false, a, /*neg_b=*/false, b,
      /*c_mod=*/(short)0, c, /*reuse_a=*/false, /*reuse_b=*/false);
}

__device__ __forceinline__ v8f zero8() {
  v8f z = {0.f, 0.f, 0.f, 0.f, 0.f, 0.f, 0.f, 0.f};
  return z;
}

// ---------------------------------------------------------------------------
// Projection GEMM: out = A[rows x Kd] @ W[Kd x 256]
//   TRANS=false : out row-major [rows x 256]                  (Q = G @ Wq)
//   TRANS=true  : out = K^T laid out [B][256][L]              (K = H @ Wk)
// 16 rows per workgroup, 8 waves x two 16x16 N-tiles = all 256 cols.
// ---------------------------------------------------------------------------
template <bool TRANS>
__global__ __launch_bounds__(256) void proj_gemm_kernel(
    const float* __restrict__ A, const float* __restrict__ W,
    float* __restrict__ out, int Kd) {
  const int tid  = threadIdx.x;
  const int w    = tid >> 5;
  const int lane = tid & 31;
  const int half = lane >> 4;
  const int l15  = lane & 15;
  const int row_base = blockIdx.x * 16;
  const int col0 = w * 32 + l15;

  // A fragment source: row m = l15, k = kk + 2*half (+1)
  const float* Ab = A + (size_t)(row_base + l15) * Kd + 2 * half;

  v8f acc0 = zero8();
  v8f acc1 = zero8();

  for (int kk = 0; kk < Kd; kk += 4) {
    v2f a;
    a.x = Ab[kk];
    a.y = Ab[kk + 1];
    // B fragment: VGPR0 = W rows (kk+2*half), VGPR1 = (kk+2*half+1), cols lane-striped
    const float* Wb = W + (size_t)(kk + 2 * half) * DP_ + col0;
    v2f b0; b0.x = Wb[0];  b0.y = Wb[DP_];
    v2f b1; b1.x = Wb[16]; b1.y = Wb[16 + DP_];
    acc0 = wmma4(a, b0, acc0);
    acc1 = wmma4(a, b1, acc1);
  }

#pragma unroll
  for (int v = 0; v < 8; ++v) {
    const int m = v + 8 * half;              // C layout: lanes0-15 M=v, lanes16-31 M=v+8
    const size_t r = (size_t)row_base + m;
    if (!TRANS) {
      out[r * DP_ + col0]      = acc0[v];
      out[r * DP_ + col0 + 16] = acc1[v];
    } else {
      const size_t bb = r >> 13;             // r / L_
      const size_t l  = r & (size_t)(L_ - 1);
      out[(bb * DP_ + col0) * L_ + l]        = acc0[v];
      out[(bb * DP_ + col0 + 16) * L_ + l]   = acc1[v];
    }
  }
}

// ---------------------------------------------------------------------------
// Fused attention pooling: per workgroup = (batch b, 16 query rows).
// Streams L in 256-wide chunks with an online (flash) softmax.
//   wave w owns: 32 logits cols in phase A, 128 DH output cols in phase C.
// ---------------------------------------------------------------------------
__global__ __launch_bounds__(256) void attn_kernel(
    const float* __restrict__ H, const float* __restrict__ Q,
    const float* __restrict__ KT, const int* __restrict__ mask,
    float* __restrict__ Z) {
  __shared__ float q_lds[16 * DP_];   // Q tile [16][256]
  __shared__ float p_lds[16 * 256];   // P chunk [16][256]
  __shared__ float red[8 * 16];       // per-wave row partials
  __shared__ float mrow[16];          // running row max
  __shared__ float srow[16];          // running row sum
  __shared__ float frow[16];          // rescale factor this chunk

  const int tid  = threadIdx.x;
  const int w    = tid >> 5;
  const int lane = tid & 31;
  const int half = lane >> 4;
  const int l15  = lane & 15;
  const int b    = blockIdx.x >> 5;          // 32 query tiles per batch
  const int t0   = (blockIdx.x & 31) * 16;

  const float* Qb = Q + ((size_t)b * T_ + t0) * DP_;
  for (int i = tid; i < 16 * DP_; i += 256) q_lds[i] = Qb[i];
  if (tid < 16) { mrow[tid] = -3.0e38f; srow[tid] = 0.f; }
  __syncthreads();

  v8f zacc[8];
#pragma unroll
  for (int nt = 0; nt < 8; ++nt) zacc[nt] = zero8();

  const float scale = 0.0625f;  // DP^-0.5
  const float* KTb = KT + (size_t)b * DP_ * L_;
  const int colw = w * 32 + l15;

  for (int lc = 0; lc < L_; lc += 256) {
    // ---- phase A: S = Q @ K^T (this wave's 32 L-columns) -----------------
    v8f s0 = zero8(), s1 = zero8();
    for (int kk = 0; kk < DP_; kk += 4) {
      v2f a;
      const int ai = l15 * DP_ + kk + 2 * half;
      a.x = q_lds[ai];
      a.y = q_lds[ai + 1];
      const float* kp = KTb + (size_t)(kk + 2 * half) * L_ + lc + colw;
      v2f b0; b0.x = kp[0];  b0.y = kp[L_];
      v2f b1; b1.x = kp[16]; b1.y = kp[16 + L_];
      s0 = wmma4(a, b0, s0);
      s1 = wmma4(a, b1, s1);
    }

    // scale + mask + per-row chunk max (rows m=v for lanes0-15, m=v+8 for 16-31)
    const int mv0 = mask[(size_t)b * L_ + lc + colw];
    const int mv1 = mask[(size_t)b * L_ + lc + colw + 16];
#pragma unroll
    for (int v = 0; v < 8; ++v) {
      float x0 = s0[v] * scale; if (mv0) x0 = -3.0e38f;
      float x1 = s1[v] * scale; if (mv1) x1 = -3.0e38f;
      s0[v] = x0; s1[v] = x1;
      float x = fmaxf(x0, x1);
      x = fmaxf(x, __shfl_xor(x, 8, 16));
      x = fmaxf(x, __shfl_xor(x, 4, 16));
      x = fmaxf(x, __shfl_xor(x, 2, 16));
      x = fmaxf(x, __shfl_xor(x, 1, 16));
      if (l15 == 0) red[w * 16 + v + 8 * half] = x;
    }
    __syncthreads();

    if (tid < 16) {
      float cm = red[tid];
#pragma unroll
      for (int ww = 1; ww < 8; ++ww) cm = fmaxf(cm, red[ww * 16 + tid]);
      const float mo = mrow[tid];
      const float mn = fmaxf(mo, cm);
      const float f  = __expf(mo - mn);
      srow[tid] *= f;
      frow[tid] = f;
      mrow[tid] = mn;
    }
    __syncthreads();

    // ---- phase B: rescale Z, P = exp(S - m), stage P, row sums -----------
    float fsel[8], msel[8];
#pragma unroll
    for (int v = 0; v < 8; ++v) {
      fsel[v] = frow[v + 8 * half];
      msel[v] = mrow[v + 8 * half];
    }
#pragma unroll
    for (int nt = 0; nt < 8; ++nt)
#pragma unroll
      for (int v = 0; v < 8; ++v) zacc[nt][v] *= fsel[v];

#pragma unroll
    for (int v = 0; v < 8; ++v) {
      const float p0 = __expf(s0[v] - msel[v]);
      const float p1 = __expf(s1[v] - msel[v]);
      const int row = v + 8 * half;
      p_lds[row * 256 + colw]      = p0;
      p_lds[row * 256 + colw + 16] = p1;
      float x = p0 + p1;
      x += __shfl_xor(x, 8, 16);
      x += __shfl_xor(x, 4, 16);
      x += __shfl_xor(x, 2, 16);
      x += __shfl_xor(x, 1, 16);
      if (l15 == 0) red[w * 16 + row] = x;
    }
    __syncthreads();

    if (tid < 16) {
      float acc = 0.f;
#pragma unroll
      for (int ww = 0; ww < 8; ++ww) acc += red[ww * 16 + tid];
      srow[tid] += acc;
    }
    __syncthreads();

    // ---- phase C: Z += P @ H_chunk (this wave's 128 DH columns) ----------
    const float* Hb = H + ((size_t)b * L_ + lc) * DH_ + w * 128 + l15;
    for (int kk = 0; kk < 256; kk += 4) {
      v2f a;
      const int ai = l15 * 256 + kk + 2 * half;
      a.x = p_lds[ai];
      a.y = p_lds[ai + 1];
      const float* hrow = Hb + (size_t)(kk + 2 * half) * DH_;
#pragma unroll
      for (int nt = 0; nt < 8; ++nt) {
        v2f bb;
        bb.x = hrow[nt * 16];
        bb.y = hrow[nt * 16 + DH_];
        zacc[nt] = wmma4(a, bb, zacc[nt]);
      }
    }
  }

  // ---- finalize: Z /= row sum -------------------------------------------
  float inv[8];
#pragma unroll
  for (int v = 0; v < 8; ++v) inv[v] = 1.0f / srow[v + 8 * half];
  float* Zb = Z + ((size_t)b * T_ + t0) * DH_;
#pragma unroll
  for (int nt = 0; nt < 8; ++nt)
#pragma unroll
    for (int v = 0; v < 8; ++v) {
      const int m = v + 8 * half;
      const int d = w * 128 + nt * 16 + l15;
      Zb[(size_t)m * DH_ + d] = zacc[nt][v] * inv[v];
    }
}

// ---------------------------------------------------------------------------
extern "C" void kernel_launch(void* const* d_in, const int* in_sizes, int n_in,
                              void* d_out, int out_size, void* d_ws, size_t ws_size,
                              hipStream_t stream) {
  const float* H    = (const float*)d_in[0];   // [B,L,DH]
  const float* G    = (const float*)d_in[1];   // [B,T,DG]
  const float* Wk   = (const float*)d_in[2];   // [DH,DP]
  const float* Wq   = (const float*)d_in[3];   // [DG,DP]
  const int*   mask = (const int*)d_in[4];     // [B,L]
  float*       Z    = (float*)d_out;           // [B,T,DH]

  float* Qws  = (float*)d_ws;                        //  4 MB: [B,T,DP]
  float* KTws = Qws + (size_t)B_ * T_ * DP_;         // 64 MB: [B,DP,L]

  // Q = G @ Wq  (row-major)
  proj_gemm_kernel<false><<<(B_ * T_) / 16, 256, 0, stream>>>(G, Wq, Qws, DG_);
  // K^T = (H @ Wk)^T  stored [B][DP][L]
  proj_gemm_kernel<true><<<(B_ * L_) / 16, 256, 0, stream>>>(H, Wk, KTws, DH_);
  // fused logits + online softmax + weighted sum of H
  attn_kernel<<<B_ * (T_ / 16), 256, 0, stream>>>(H, Qws, KTws, mask, Z);
}